// OBBNMSAndReturnAsBatchedResult_23716809409209
// MI455X (gfx1250) — compile-verified
//
#include <hip/hip_runtime.h>
#include <hip/hip_bf16.h>
#include <stdint.h>

// ---------------------------------------------------------------------------
// OBB Matrix-NMS (ProbIoU, gaussian decay) for MI455X / gfx1250.
//
// B=32, N=1024, C=15, M=300. conf_t=0.1, iou_t=0.2, sigma=3, already sorted.
//
// Pipeline:
//   k_prep  : per-box GBB features, pre-packed in WMMA B-fragment lane order
//   k_pair<1>: cmax[j]^2 = (max_{i<j} iou)^2  via WMMA tiles     (B x 4 x 256)
//   k_pair<2>: M[j] = max_{i<j}(iou^2 - cmax_i^2) via WMMA tiles (B x 4 x 256)
//   k_final : decay=exp(-M/3), keep, wave32 ballot-scan compaction (B x 1024)
//
// Pairwise bilinear terms (denom rank-5, num1 rank-10, num2 rank-8) are
// chained V_WMMA_F32_16X16X4_F32. Per-box features are stored in a 144-byte
// record whose layout IS the B-fragment register layout (h=0 K-pairs at +0,
// h=1 K-pairs at +64B), so the inner loop builds B fragments with three
// ds_load_b128 + one ds_load_b64 and zero selects. The record table is staged
// global->LDS with async b128 DMA (s_wait_asynccnt fence).
// Coordinates are scaled by 1/32 and centered (ProbIoU is scale-invariant).
// ---------------------------------------------------------------------------

#define NB 32
#define NN 1024
#define NC 15
#define NM 300
#define EPSF 1e-7f
#define NREC 36  // floats per box record (144 B, 16B-aligned, bank-friendly)

typedef __attribute__((ext_vector_type(2))) float v2f;
typedef __attribute__((ext_vector_type(8))) float v8f;

// workspace layout (float units)
#define WS_VREC 0                          // 32768 * 36 floats (4.7 MB)
#define WS_CONF (NB * NN * NREC)
#define WS_LAB  (WS_CONF + NB * NN)
#define WS_CMX2 (WS_LAB + NB * NN)
#define WS_MJ   (WS_CMX2 + NB * NN)

static __device__ __forceinline__ float fast_rcp(float x) {
  return __builtin_amdgcn_rcpf(x);
}
static __device__ __forceinline__ float fast_sqrt(float x) {
  return __builtin_amdgcn_sqrtf(x);
}

// --------------------------- k_prep ---------------------------------------
// v-feature order (rank-K factor columns), k = 0..27:
//  denom: v0=b v1=a v2=1 v3=dr v4=-2c  (k5..7 = 0)
//  num1 : v8=1 v9=y v10=yy v11=a v12=ay v13=q v14=x v15=xx v16=b v17=bx
//  num2 : v20=x v21=xy v22=1 v23=y v24=cx v25=-cxy v26=c v27=cy
// record: [0..13]  = h0 pairs (v0,v1)(v4,v5)(v8,v9)(v12,v13)(v16,v17)(v20,v21)(v24,v25)
//         [14..15] = pad
//         [16..29] = h1 pairs (v2,v3)(v6,v7)(v10,v11)(v14,v15)(v18,v19)(v22,v23)(v26,v27)
//         [30]     = s = sqrt(max(det,0)) ; rest pad
__global__ __launch_bounds__(256) void k_prep(const float* __restrict__ boxes,
                                              const float* __restrict__ scores,
                                              float* __restrict__ ws) {
  const int g = blockIdx.x * 256 + threadIdx.x;  // 0 .. B*N-1
  const float* bp = boxes + (size_t)g * 5;
  float cx0 = bp[0], cy0 = bp[1], w = bp[2], h = bp[3], r = bp[4];

  const float* sp = scores + (size_t)g * NC;
  float best = sp[0];
  int lbl = 0;
#pragma unroll
  for (int k = 1; k < NC; ++k) {
    float s = sp[k];
    if (s > best) { best = s; lbl = k; }
  }
  float conf = (best < 0.1f) ? 0.0f : best;

  // similarity transform (iou-invariant): scale 1/32, center
  const float SCL = 1.0f / 32.0f;
  float x = cx0 * SCL - 16.0f;
  float y = cy0 * SCL - 16.0f;
  float ww = w * SCL, hh = h * SCL;
  float ea = ww * ww * (1.0f / 12.0f);
  float eb = hh * hh * (1.0f / 12.0f);
  float cs = __cosf(r), sn = __sinf(r);
  float va = ea * cs * cs + eb * sn * sn;  // var_x  ("a")
  float vb = ea * sn * sn + eb * cs * cs;  // var_y  ("b")
  float vc = (ea - eb) * cs * sn;          // cov_xy ("c")
  float dr = fmaf(va, vb, -vc * vc);       // raw det (for denom)
  float s2 = fast_sqrt(fmaxf(dr, 0.0f));   // sqrt(clipped det), for t3

  float xx = x * x, yy = y * y, xy = x * y;
  float ay = va * y, bx = vb * x, cxv = vc * x, cyv = vc * y;
  float q = fmaf(va, yy, vb * xx);
  float mcxy = -(vc * xy);

  float4* rw = (float4*)(ws + WS_VREC + (size_t)g * NREC);
  rw[0] = {vb, va, -2.0f * vc, 0.0f};   // v0,v1,v4,v5
  rw[1] = {1.0f, y, ay, q};             // v8,v9,v12,v13
  rw[2] = {vb, bx, x, xy};              // v16,v17,v20,v21
  rw[3] = {cxv, mcxy, 0.0f, 0.0f};      // v24,v25, pad
  rw[4] = {1.0f, dr, 0.0f, 0.0f};       // v2,v3,v6,v7
  rw[5] = {yy, va, x, xx};              // v10,v11,v14,v15
  rw[6] = {0.0f, 0.0f, 1.0f, y};        // v18,v19,v22,v23
  rw[7] = {vc, cyv, s2, 0.0f};          // v26,v27, s, pad
  rw[8] = {0.0f, 0.0f, 0.0f, 0.0f};
  ws[WS_CONF + g] = conf;
  ((int*)ws)[WS_LAB + g] = lbl;
}

// ------------------------- WMMA pair tile pass -----------------------------
// One wave owns one 16-wide j-tile; loops over 16-wide i-chunks (i<j).
// D[m][n] = sum_k u_k(J_{jbase+m}) * v_k(I_{ibase+n}) via v_wmma_f32_16x16x4.
// PHASE 1: track max iou per j (cmax). PHASE 2: track max(iou^2 - cmax_i^2).
template <int PHASE>
__device__ __forceinline__ void tile_pass(const float* __restrict__ ldsV,
                                          const float* __restrict__ ldsCmax2,
                                          float* __restrict__ wsOut, int tile) {
  const int lane = threadIdx.x & 31;
  const int h = lane >> 4;  // lane-half selects the K pair (ISA A/B layout)
  const int n = lane & 15;  // matrix row (A) / column (B,D)
  const int jbase = tile << 4;

  // ---- A fragments (loop-invariant per tile): u-features of box J ----
  const float* rj = ldsV + (size_t)(jbase + n) * NREC;
  float4 q0 = *(const float4*)(rj + 0);    // b, a, -2c, 0
  float4 q1 = *(const float4*)(rj + 4);    // 1, y, ay, q
  float4 q2 = *(const float4*)(rj + 8);    // b, bx, x, xy
  float4 q3 = *(const float4*)(rj + 12);   // cx, -cxy, pad, pad
  float4 q4 = *(const float4*)(rj + 16);   // 1, dr, 0, 0
  float4 q5 = *(const float4*)(rj + 20);   // yy, a, x, xx
  float4 q7 = *(const float4*)(rj + 28);   // c, cy, s, pad
  float a = q0.y, b = q0.x, y = q1.y, ay = q1.z, qq = q1.w;
  float bx = q2.y, x = q2.z, xy = q2.w, cxv = q3.x, mcxy = q3.y;
  float dr = q4.y, yy = q5.x, xx = q5.w, c = q7.x, cyv = q7.y;

  v2f Af[7];
  Af[0].x = h ? dr : a;            Af[0].y = h ? 1.0f : b;
  Af[1].x = h ? 0.0f : c;          Af[1].y = 0.0f;
  Af[2].x = h ? a : qq;            Af[2].y = h ? yy : -2.0f * ay;
  Af[3].x = h ? -2.0f * bx : -2.0f * y;  Af[3].y = h ? b : 1.0f;
  Af[4].x = h ? 0.0f : xx;         Af[4].y = h ? 0.0f : -2.0f * x;
  Af[5].x = h ? mcxy : cyv;        Af[5].y = h ? cxv : -c;
  Af[6].x = h ? -xy : y;           Af[6].y = h ? x : 1.0f;

  float sJ[8];
#pragma unroll
  for (int v = 0; v < 8; ++v)
    sJ[v] = ldsV[(size_t)(jbase + v + 8 * h) * NREC + 30];

  float mx[8];
#pragma unroll
  for (int v = 0; v < 8; ++v) mx[v] = 0.0f;

  for (int ibase = 0; ibase <= jbase; ibase += 16) {
    // ---- B fragments: direct ds loads, pre-packed in lane order ----
    const float* ri = ldsV + (size_t)(ibase + n) * NREC + h * 16;
    float4 B0 = *(const float4*)(ri + 0);
    float4 B1 = *(const float4*)(ri + 4);
    float4 B2 = *(const float4*)(ri + 8);
    float2 BP = *(const float2*)(ri + 12);
    float sI = ldsV[(size_t)(ibase + n) * NREC + 30];
    float c2I = 0.0f;
    if (PHASE == 2) c2I = ldsCmax2[ibase + n];

    v2f Bf[7];
    Bf[0].x = B0.x; Bf[0].y = B0.y;
    Bf[1].x = B0.z; Bf[1].y = B0.w;
    Bf[2].x = B1.x; Bf[2].y = B1.y;
    Bf[3].x = B1.z; Bf[3].y = B1.w;
    Bf[4].x = B2.x; Bf[4].y = B2.y;
    Bf[5].x = B2.z; Bf[5].y = B2.w;
    Bf[6].x = BP.x; Bf[6].y = BP.y;

    v8f den = {0.f, 0.f, 0.f, 0.f, 0.f, 0.f, 0.f, 0.f};
    v8f n1 = {0.f, 0.f, 0.f, 0.f, 0.f, 0.f, 0.f, 0.f};
    v8f n2 = {0.f, 0.f, 0.f, 0.f, 0.f, 0.f, 0.f, 0.f};
    den = __builtin_amdgcn_wmma_f32_16x16x4_f32(false, Af[0], false, Bf[0],
                                                (short)0, den, false, false);
    den = __builtin_amdgcn_wmma_f32_16x16x4_f32(false, Af[1], false, Bf[1],
                                                (short)0, den, false, false);
    n1 = __builtin_amdgcn_wmma_f32_16x16x4_f32(false, Af[2], false, Bf[2],
                                               (short)0, n1, false, false);
    n1 = __builtin_amdgcn_wmma_f32_16x16x4_f32(false, Af[3], false, Bf[3],
                                               (short)0, n1, false, false);
    n1 = __builtin_amdgcn_wmma_f32_16x16x4_f32(false, Af[4], false, Bf[4],
                                               (short)0, n1, false, false);
    n2 = __builtin_amdgcn_wmma_f32_16x16x4_f32(false, Af[5], false, Bf[5],
                                               (short)0, n2, false, false);
    n2 = __builtin_amdgcn_wmma_f32_16x16x4_f32(false, Af[6], false, Bf[6],
                                               (short)0, n2, false, false);

    const bool diag = (ibase == jbase);
#pragma unroll
    for (int v = 0; v < 8; ++v) {
      float dsum = den[v] + EPSF;
      float rd = fast_rcp(dsum);
      float t12 = rd * fmaf(0.25f, n1[v], 0.5f * n2[v]);
      float t3 =
          0.5f * __logf(fmaf(dsum, fast_rcp(fmaf(4.0f, sJ[v] * sI, EPSF)), EPSF));
      float bd = fminf(fmaxf(t12 + t3, EPSF), 100.0f);
      float iou = 1.0f - fast_sqrt(1.0f - __expf(-bd) + EPSF);
      float val = (PHASE == 1) ? iou : fmaf(iou, iou, -c2I);
      bool valid = !diag || (n < v + 8 * h);  // need i < j
      val = valid ? val : 0.0f;               // masked entries contribute 0
      mx[v] = fmaxf(mx[v], val);
    }
  }

  // reduce across the 16 lanes of each half (j rows 0-7 -> lane 0, 8-15 -> 16)
#pragma unroll
  for (int v = 0; v < 8; ++v) {
    float m = mx[v];
    m = fmaxf(m, __shfl_xor(m, 1));
    m = fmaxf(m, __shfl_xor(m, 2));
    m = fmaxf(m, __shfl_xor(m, 4));
    m = fmaxf(m, __shfl_xor(m, 8));
    mx[v] = m;
  }
  if (n == 0) {
#pragma unroll
    for (int v = 0; v < 8; ++v) {
      int j = jbase + v + 8 * h;
      if (PHASE == 1)
        wsOut[j] = mx[v] * mx[v];  // store cmax^2
      else
        wsOut[j] = mx[v];          // store M_j
    }
  }
}

// --------------------------- k_pair ----------------------------------------
template <int PHASE>
__global__ __launch_bounds__(256) void k_pair(float* __restrict__ ws) {
  __shared__ __align__(16) float ldsV[NN * NREC];  // 144 KB record table
  __shared__ __align__(16) float ldsC[NN];         // cmax^2 (phase 2)
  const int b = blockIdx.x;
  const int t = threadIdx.x;

  // Stage the batch's record table with CDNA5 async global->LDS DMA.
  {
    const float* gV = ws + WS_VREC + (size_t)b * NN * NREC;
    unsigned ldsBase = (unsigned)(uintptr_t)(&ldsV[0]);
#pragma unroll
    for (int k = 0; k < 36; ++k) {  // 256 thr * 576 B = 147456 B
      unsigned off = (unsigned)t * 576u + (unsigned)k * 16u;
      asm volatile(
          "global_load_async_to_lds_b128 %0, %1, %2 offset:0" ::"v"(ldsBase +
                                                                    off),
          "v"(off), "s"(gV)
          : "memory");
    }
  }
  if (PHASE == 2) {
    const float* gC = ws + WS_CMX2 + (size_t)b * NN;
    unsigned ldsDst = (unsigned)(uintptr_t)(&ldsC[0]) + (unsigned)t * 16u;
    unsigned goff = (unsigned)t * 16u;
    asm volatile(
        "global_load_async_to_lds_b128 %0, %1, %2 offset:0" ::"v"(ldsDst),
        "v"(goff), "s"(gC)
        : "memory");
  }
  asm volatile("s_wait_asynccnt 0" ::: "memory");
  __syncthreads();

  const int wv = t >> 5;
  const int gw = blockIdx.y * 8 + wv;  // 0..31 over (blockIdx.y in [0,4))
  float* outArr = (PHASE == 1) ? (ws + WS_CMX2 + (size_t)b * NN)
                               : (ws + WS_MJ + (size_t)b * NN);
  // balanced tile pair: (gw) + (63-gw) -> 65 i-chunks per wave
  tile_pass<PHASE>(ldsV, ldsC, outArr, gw);
  tile_pass<PHASE>(ldsV, ldsC, outArr, 63 - gw);
}

// --------------------------- k_final ---------------------------------------
__global__ __launch_bounds__(1024) void k_final(const float* __restrict__ boxes,
                                                const float* __restrict__ ws,
                                                float* __restrict__ out) {
  __shared__ int WTOT[32];
  __shared__ int SEL[NM];
  const int b = blockIdx.x, tid = threadIdx.x;
  const int g = b * NN + tid;

  float Mj = ws[WS_MJ + g];
  float conf = ws[WS_CONF + g];
  float decay = __expf(-Mj * (1.0f / 3.0f));  // min_i exp(.) == exp(-max/sigma)
  bool keep = (conf * decay) > 0.2f;

  unsigned long long bal = __ballot(keep);  // wave32: low 32 bits
  int lane = tid & 31, wv = tid >> 5;
  int pfx = __popcll(bal & ((1ull << lane) - 1ull));
  if (lane == 0) WTOT[wv] = __popcll(bal);
  if (tid < NM) SEL[tid] = -1;
  __syncthreads();

  int base = 0, tot = 0;
#pragma unroll
  for (int w2 = 0; w2 < 32; ++w2) {
    int c = WTOT[w2];
    tot += c;
    if (w2 < wv) base += c;
  }
  if (keep) {
    int r2 = base + pfx;
    if (r2 < NM) SEL[r2] = tid;  // stable: rank increases with tid
  }
  __syncthreads();

  int num = tot < NM ? tot : NM;
  if (tid == 0) out[b] = (float)num;
  if (tid < NM) {
    int s = SEL[tid];
    float* bo = out + NB + ((size_t)b * NM + tid) * 5;
    float* so = out + NB + (size_t)NB * NM * 5 + (size_t)b * NM + tid;
    float* co = so + (size_t)NB * NM;
    if (s >= 0) {
      const float* gb = boxes + ((size_t)b * NN + s) * 5;
      bo[0] = gb[0]; bo[1] = gb[1]; bo[2] = gb[2]; bo[3] = gb[3]; bo[4] = gb[4];
      *so = ws[WS_CONF + b * NN + s];
      *co = (float)((const int*)ws)[WS_LAB + b * NN + s];
    } else {
      bo[0] = bo[1] = bo[2] = bo[3] = bo[4] = -1.0f;
      *so = -1.0f;
      *co = -1.0f;
    }
  }
}

// --------------------------- launch ----------------------------------------
extern "C" void kernel_launch(void* const* d_in, const int* in_sizes, int n_in,
                              void* d_out, int out_size, void* d_ws,
                              size_t ws_size, hipStream_t stream) {
  const float* boxes = (const float*)d_in[0];   // [32,1024,5] f32
  const float* scores = (const float*)d_in[1];  // [32,1024,15] f32
  float* out = (float*)d_out;                   // 67232 f32
  float* ws = (float*)d_ws;                     // ~5.3 MB used

  k_prep<<<dim3((NB * NN) / 256), 256, 0, stream>>>(boxes, scores, ws);
  k_pair<1><<<dim3(NB, 4), 256, 0, stream>>>(ws);
  k_pair<2><<<dim3(NB, 4), 256, 0, stream>>>(ws);
  k_final<<<dim3(NB), 1024, 0, stream>>>(boxes, ws, out);
}